// ClusteringSeedSelection_26061861552501
// MI455X (gfx1250) — compile-verified
//
#include <hip/hip_runtime.h>
#include <math.h>

// ---------------------------------------------------------------------------
// ClusteringSeedSelection for MI455X (gfx1250, wave32, WMMA)
//   B=16, Q=1800, C=256.  Gram matrix on V_WMMA_F32_16X16X4_F32 (exact f32),
//   upper-triangular tiles only (symmetric), adjacency as L2-resident
//   bitsets, CC labels + compaction per batch.
// ---------------------------------------------------------------------------

#define BATCH 16
#define Q     1800
#define C     256
#define GT    29            // ceil(1800/64) 64-row block tiles
#define NTRI  (GT * (GT + 1) / 2)   // 435 upper-triangular block tiles
#define WQ    57            // ceil(1800/32) adjacency words per row
#define SW    64            // padded word stride (per row / per sel row)
#define KC    64            // K chunk staged in LDS
#define LDA   66            // padded LDS row stride (floats) -> conflict-free

typedef __attribute__((ext_vector_type(2))) float v2f;
typedef __attribute__((ext_vector_type(8))) float v8f;

// ---------------------------------------------------------------------------
// Kernel 0: zero the adjacency bitset (written with atomicOr later)
// ---------------------------------------------------------------------------
__global__ void zero_kernel(unsigned* __restrict__ p, size_t n) {
    size_t i = (size_t)blockIdx.x * blockDim.x + threadIdx.x;
    if (i < n) p[i] = 0u;
}

// ---------------------------------------------------------------------------
// Kernel 1: scores = sigmoid(logits); selected bitset (>=0.5) with argmax
// fallback when a batch selects nothing.  One block per batch.
// ---------------------------------------------------------------------------
__global__ __launch_bounds__(1024)
void score_kernel(const float* __restrict__ logits,
                  float* __restrict__ scores,
                  unsigned* __restrict__ selbits) {
    __shared__ float rs[1024];
    __shared__ int   ri[1024];
    __shared__ unsigned anyw;
    const int b = blockIdx.x;
    const int tid = threadIdx.x;
    if (tid == 0) anyw = 0u;
    __syncthreads();

    float bests = -1.0f; int besti = 0;
    for (int r = 0; r < 2; ++r) {
        int q = tid + r * 1024;
        bool pred = false;
        if (q < Q) {
            float x = logits[b * Q + q];
            float s = 1.0f / (1.0f + expf(-x));
            scores[b * Q + q] = s;
            pred = (s >= 0.5f);
            if (s > bests) { bests = s; besti = q; }   // q ascends per thread
        }
        unsigned m = (unsigned)__ballot(pred);          // wave32: low 32 bits
        int w = q >> 5;                                 // uniform per wave
        if ((tid & 31) == 0 && w < SW) {
            selbits[b * SW + w] = m;
            if (m) atomicOr(&anyw, 1u);
        }
    }
    rs[tid] = bests; ri[tid] = besti;
    __syncthreads();
    for (int s2 = 512; s2 > 0; s2 >>= 1) {
        if (tid < s2) {
            bool take = (rs[tid + s2] > rs[tid]) ||
                        (rs[tid + s2] == rs[tid] && ri[tid + s2] < ri[tid]);
            if (take) { rs[tid] = rs[tid + s2]; ri[tid] = ri[tid + s2]; }
        }
        __syncthreads();
    }
    if (tid == 0 && anyw == 0u) {
        int fb = ri[0];
        selbits[b * SW + (fb >> 5)] = 1u << (fb & 31);  // batch had no bits set
    }
}

// ---------------------------------------------------------------------------
// Kernel 2: inverse row norms (one wave per signature row)
// ---------------------------------------------------------------------------
__global__ __launch_bounds__(256)
void norm_kernel(const float* __restrict__ sig, float* __restrict__ invn) {
    const int tid = threadIdx.x, lane = tid & 31, wv = tid >> 5;
    const int row = blockIdx.x * 8 + wv;                // < BATCH*Q = 28800
    const float* p = sig + (size_t)row * C + lane * 8;
    float4 a = *(const float4*)p;
    float4 c = *(const float4*)(p + 4);
    float ss = a.x*a.x + a.y*a.y + a.z*a.z + a.w*a.w
             + c.x*c.x + c.y*c.y + c.z*c.z + c.w*c.w;
    for (int off = 16; off; off >>= 1) ss += __shfl_xor(ss, off, 32);
    if (lane == 0) invn[row] = 1.0f / fmaxf(sqrtf(ss), 1e-12f);
}

// ---------------------------------------------------------------------------
// Kernel 3: Gram tiles via V_WMMA_F32_16X16X4_F32 over upper-triangular
// block tiles only; threshold in-register into adjacency bitsets, emitting
// both (i->j) and (j->i) directions for off-diagonal tiles.
// Block = 256 thr = 8 waves computes a 64x64 tile.
// ---------------------------------------------------------------------------

// Row-major emission: adjacency bits for rows i (I range) over columns j.
__device__ __forceinline__
void emit_tile(v8f acc, int b, int ibase, int jbase,
               const unsigned* __restrict__ selbits,
               unsigned* __restrict__ adj) {
    const int lane = threadIdx.x & 31;
    // 16 selection bits for columns jbase..jbase+15 (jbase is 16-aligned)
    const unsigned selj =
        (selbits[b * SW + (jbase >> 5)] >> (jbase & 31)) & 0xFFFFu;
    const int rowoff = (lane >> 4) * 8;                 // 0 or 8
#pragma unroll
    for (int v = 0; v < 8; ++v) {
        bool pred = (acc[v] >= 0.8f) && (((selj >> (lane & 15)) & 1u) != 0u);
        unsigned m = (unsigned)__ballot(pred);
        // lanes 0..15 carry row ibase+v, lanes 16..31 carry row ibase+v+8
        int i = ibase + v + rowoff;
        unsigned bits16 = rowoff ? (m >> 16) : (m & 0xFFFFu);
        if ((lane & 15) == 0 && i < Q) {
            unsigned seli = (selbits[b * SW + (i >> 5)] >> (i & 31)) & 1u;
            if (seli && bits16)
                atomicOr(&adj[((size_t)b * Q + i) * SW + (jbase >> 5)],
                         bits16 << (jbase & 31));
        }
    }
}

// Column-major (transposed) emission: adjacency bits for rows j (J range)
// over columns i.  Each lane holds a column of the 16x16 tile, so no bit
// transpose is needed: accumulate 8 bits, OR with the mirror half-wave.
__device__ __forceinline__
void emit_tile_T(v8f acc, int b, int ibase, int jbase,
                 const unsigned* __restrict__ selbits,
                 unsigned* __restrict__ adj) {
    const int lane = threadIdx.x & 31;
    // 16 selection bits for rows ibase..ibase+15 (bit positions of column i)
    const unsigned seli16 =
        (selbits[b * SW + (ibase >> 5)] >> (ibase & 31)) & 0xFFFFu;
    const int rowoff = (lane >> 4) * 8;                 // 0 or 8
    unsigned cb = 0u;
#pragma unroll
    for (int v = 0; v < 8; ++v) {
        bool pred = (acc[v] >= 0.8f) &&
                    (((seli16 >> (rowoff + v)) & 1u) != 0u);
        cb |= (pred ? 1u : 0u) << (rowoff + v);
    }
    cb |= (unsigned)__shfl_xor((int)cb, 16, 32);        // merge half-columns
    int j = jbase + (lane & 15);
    if ((lane >> 4) == 0 && j < Q) {
        unsigned selj = (selbits[b * SW + (j >> 5)] >> (j & 31)) & 1u;
        if (selj && cb)
            atomicOr(&adj[((size_t)b * Q + j) * SW + (ibase >> 5)],
                     cb << (ibase & 31));
    }
}

__global__ __launch_bounds__(256)
void gram_adj_kernel(const float* __restrict__ sig,
                     const float* __restrict__ invn,
                     const unsigned* __restrict__ selbits,
                     unsigned* __restrict__ adj) {
    __shared__ float lA[64][LDA];
    __shared__ float lB[64][LDA];
    // decode triangular block index (scalar, uniform): tile (I,J), I<=J
    int rem = blockIdx.x;
    int I = 0;
    while (rem >= GT - I) { rem -= GT - I; ++I; }
    const int J  = I + rem;
    const int b  = blockIdx.y;
    const int i0 = I * 64;
    const int j0 = J * 64;

    const int tid  = threadIdx.x;
    const int lane = tid & 31;
    const int wv   = tid >> 5;                          // 0..7
    // each wave owns subtiles (ti, tj0) and (ti, tj0+1): A fragment shared
    const int ti  = wv >> 1;
    const int tj0 = (wv & 1) * 2;

    v8f acc0 = {}; v8f acc1 = {};
    const int lr = lane & 15;
    const int kb = (lane >> 4) * 2;                     // fragment k base

    for (int c0 = 0; c0 < C; c0 += KC) {
        // ---- stage 64x64 chunks of A and B rows, pre-scaled by 1/||row|| --
        {
            const int r  = tid >> 2;                    // 0..63
            const int qo = (tid & 3) * 16;              // col offset in chunk
            int gi = i0 + r;
            if (gi < Q) {
                float s = invn[b * Q + gi];
                const float4* p = (const float4*)(sig + ((size_t)b * Q + gi) * C + c0 + qo);
#pragma unroll
                for (int t = 0; t < 4; ++t) {
                    float4 v = p[t];
                    lA[r][qo + t*4 + 0] = v.x * s;
                    lA[r][qo + t*4 + 1] = v.y * s;
                    lA[r][qo + t*4 + 2] = v.z * s;
                    lA[r][qo + t*4 + 3] = v.w * s;
                }
            } else {
#pragma unroll
                for (int t = 0; t < 16; ++t) lA[r][qo + t] = 0.0f;
            }
            int gj = j0 + r;
            if (gj < Q) {
                float s = invn[b * Q + gj];
                const float4* p = (const float4*)(sig + ((size_t)b * Q + gj) * C + c0 + qo);
#pragma unroll
                for (int t = 0; t < 4; ++t) {
                    float4 v = p[t];
                    lB[r][qo + t*4 + 0] = v.x * s;
                    lB[r][qo + t*4 + 1] = v.y * s;
                    lB[r][qo + t*4 + 2] = v.z * s;
                    lB[r][qo + t*4 + 3] = v.w * s;
                }
            } else {
#pragma unroll
                for (int t = 0; t < 16; ++t) lB[r][qo + t] = 0.0f;
            }
        }
        __syncthreads();

        // ---- WMMA over the K chunk.  A 16x4 and B 4x16 f32 fragments have
        // identical per-lane addressing: row = lane&15, k = kb + {0,1}.
        const float* arow  = &lA[ti * 16 + lr][kb];
        const float* brow0 = &lB[tj0 * 16 + lr][kb];
        const float* brow1 = &lB[(tj0 + 1) * 16 + lr][kb];
#pragma unroll
        for (int k = 0; k < KC; k += 4) {
            v2f a  = *(const v2f*)(arow  + k);
            v2f b0 = *(const v2f*)(brow0 + k);
            v2f b1 = *(const v2f*)(brow1 + k);
            acc0 = __builtin_amdgcn_wmma_f32_16x16x4_f32(
                false, a, false, b0, (short)0, acc0, false, false);
            acc1 = __builtin_amdgcn_wmma_f32_16x16x4_f32(
                false, a, false, b1, (short)0, acc1, false, false);
        }
        __syncthreads();
    }

    const int ib  = i0 + ti * 16;
    const int jb0 = j0 + tj0 * 16;
    const int jb1 = j0 + (tj0 + 1) * 16;
    emit_tile(acc0, b, ib, jb0, selbits, adj);
    emit_tile(acc1, b, ib, jb1, selbits, adj);
    if (I != J) {                       // off-diagonal: emit transpose too
        emit_tile_T(acc0, b, ib, jb0, selbits, adj);
        emit_tile_T(acc1, b, ib, jb1, selbits, adj);
    }
}

// ---------------------------------------------------------------------------
// Kernel 4: per-batch connected components (min-label propagation in LDS),
// per-component argmax via packed u64 ds_max, compaction.  1 block / batch.
// ---------------------------------------------------------------------------
__global__ __launch_bounds__(1024)
void cc_kernel(const unsigned* __restrict__ adj,
               const unsigned* __restrict__ selbits,
               const float* __restrict__ scores,
               unsigned* __restrict__ comp_idx,
               float* __restrict__ comp_score,
               unsigned* __restrict__ num_seeds) {
    __shared__ int labels[Q];
    __shared__ unsigned long long best[Q];
    __shared__ unsigned selw[SW];
    __shared__ unsigned actw[SW];
    __shared__ unsigned wpref[SW + 1];
    __shared__ int changed;

    const int b = blockIdx.x;
    const int tid = threadIdx.x;

    for (int i = tid; i < Q; i += 1024) { labels[i] = i; best[i] = 0ull; }
    for (int w = tid; w < SW; w += 1024) selw[w] = selbits[b * SW + w];
    __syncthreads();

    // ---- min-label propagation to fixpoint -------------------------------
    for (;;) {
        __syncthreads();
        if (tid == 0) changed = 0;
        __syncthreads();
        for (int i = tid; i < Q; i += 1024) {
            int m = labels[i];
            const unsigned* row = adj + ((size_t)b * Q + i) * SW;
            for (int w = 0; w < WQ; ++w) {
                unsigned bits = row[w];
                while (bits) {
                    int j = (w << 5) + __builtin_ctz(bits);
                    bits &= bits - 1u;
                    int lj = labels[j];
                    if (lj < m) m = lj;
                }
            }
            if (m < labels[i]) { labels[i] = m; changed = 1; }
        }
        __syncthreads();
        if (!changed) break;
    }

    // ---- per-component best: key = (score_bits<<32) | (~node) ------------
    for (int j = tid; j < Q; j += 1024) {
        if ((selw[j >> 5] >> (j & 31)) & 1u) {
            int l = labels[j];
            unsigned sb = __float_as_uint(scores[b * Q + j]);   // score>0 -> ordered
            unsigned long long key =
                ((unsigned long long)sb << 32) | (0xFFFFFFFFu - (unsigned)j);
            atomicMax(&best[l], key);
        }
    }
    __syncthreads();

    // ---- active bitset + compaction --------------------------------------
    for (int r = 0; r < 2; ++r) {
        int node = tid + r * 1024;
        bool pred = (node < Q) &&
                    (((selw[node >> 5] >> (node & 31)) & 1u) != 0u) &&
                    (labels[node] == node);
        unsigned m = (unsigned)__ballot(pred);
        int w = node >> 5;                      // uniform within a wave
        if ((tid & 31) == 0 && w < SW) actw[w] = m;
    }
    __syncthreads();
    if (tid == 0) {
        unsigned s = 0;
        for (int w = 0; w < SW; ++w) { wpref[w] = s; s += __popc(actw[w]); }
        wpref[SW] = s;
        num_seeds[b] = s;
    }
    __syncthreads();
    for (int i = tid; i < Q; i += 1024) {
        if ((((selw[i >> 5] >> (i & 31)) & 1u) != 0u) && labels[i] == i) {
            unsigned pos = wpref[i >> 5] +
                           __popc(actw[i >> 5] & ((1u << (i & 31)) - 1u));
            unsigned long long k = best[i];
            comp_idx[b * Q + pos]   = 0xFFFFFFFFu - (unsigned)(k & 0xFFFFFFFFu);
            comp_score[b * Q + pos] = __uint_as_float((unsigned)(k >> 32));
        }
    }
}

// ---------------------------------------------------------------------------
// Kernel 5: scatter outputs (signatures, mask, scores) with zero padding.
// One 64-thread block per output row.
// ---------------------------------------------------------------------------
__global__ __launch_bounds__(64)
void write_kernel(const float* __restrict__ sig,
                  const unsigned* __restrict__ comp_idx,
                  const float* __restrict__ comp_score,
                  const unsigned* __restrict__ num_seeds,
                  float* __restrict__ out_sig,
                  float* __restrict__ out_mask,
                  float* __restrict__ out_score) {
    const int blk = blockIdx.x;                 // b*Q + d
    const int b = blk / Q, d = blk % Q;
    const int tid = threadIdx.x;
    const unsigned n = num_seeds[b];
    float4 v = {0.f, 0.f, 0.f, 0.f};
    if ((unsigned)d < n) {
        unsigned j = comp_idx[blk];
        v = *(const float4*)(sig + ((size_t)b * Q + j) * C + tid * 4);
    }
    *(float4*)(out_sig + (size_t)blk * C + tid * 4) = v;
    if (tid == 0) {
        bool act = (unsigned)d < n;
        out_mask[blk]  = act ? 1.0f : 0.0f;
        out_score[blk] = act ? comp_score[blk] : 0.0f;
    }
}

// ---------------------------------------------------------------------------
extern "C" void kernel_launch(void* const* d_in, const int* in_sizes, int n_in,
                              void* d_out, int out_size, void* d_ws, size_t ws_size,
                              hipStream_t stream) {
    const float* sig    = (const float*)d_in[0];   // [16,1800,256] f32
    const float* logits = (const float*)d_in[1];   // [16,1800]     f32
    (void)in_sizes; (void)n_in; (void)out_size; (void)ws_size;

    float* out = (float*)d_out;
    float* out_sig   = out;                               // [16,1800,256]
    float* out_mask  = out + (size_t)BATCH * Q * C;       // [16,1800]
    float* out_score = out_mask + (size_t)BATCH * Q;      // [16,1800]

    char* ws = (char*)d_ws;
    float*    scores     = (float*)ws;    ws += (size_t)BATCH * Q * 4;
    float*    invn       = (float*)ws;    ws += (size_t)BATCH * Q * 4;
    unsigned* selbits    = (unsigned*)ws; ws += (size_t)BATCH * SW * 4;
    unsigned* adj        = (unsigned*)ws; ws += (size_t)BATCH * Q * SW * 4;
    unsigned* comp_idx   = (unsigned*)ws; ws += (size_t)BATCH * Q * 4;
    float*    comp_score = (float*)ws;    ws += (size_t)BATCH * Q * 4;
    unsigned* nseeds     = (unsigned*)ws; ws += (size_t)BATCH * 4;

    const size_t nadj = (size_t)BATCH * Q * SW;
    zero_kernel<<<(unsigned)((nadj + 1023) / 1024), 1024, 0, stream>>>(adj, nadj);
    score_kernel<<<BATCH, 1024, 0, stream>>>(logits, scores, selbits);
    norm_kernel<<<(BATCH * Q) / 8, 256, 0, stream>>>(sig, invn);
    dim3 g(NTRI, BATCH);
    gram_adj_kernel<<<g, 256, 0, stream>>>(sig, invn, selbits, adj);
    cc_kernel<<<BATCH, 1024, 0, stream>>>(adj, selbits, scores,
                                          comp_idx, comp_score, nseeds);
    write_kernel<<<BATCH * Q, 64, 0, stream>>>(sig, comp_idx, comp_score, nseeds,
                                               out_sig, out_mask, out_score);
}